// ocnn_37048387895734
// MI455X (gfx1250) — compile-verified
//
#include <hip/hip_runtime.h>
#include <hip/hip_bf16.h>

typedef __attribute__((ext_vector_type(2))) float v2f;
typedef __attribute__((ext_vector_type(8))) float v8f;

__device__ __forceinline__ v8f wmma_f32(v2f a, v2f b, v8f c) {
  // 8 args: (neg_a, A, neg_b, B, c_mod, C, reuse_a, reuse_b)
  return __builtin_amdgcn_wmma_f32_16x16x4_f32(false, a, false, b, (short)0, c,
                                               false, false);
}

// One wave32 per batch. Computes the closed-form of the FFT-based optical MVM:
//   Xhat[u,j] = sum_i x_cnn[i,j] e^{-2pi i * i*u/100},  u_m = 2m-50
//   S[m,k]    = sum_j Xhat[u_m,j]*|W[m,j]| e^{-2pi i * j*38(k+1)/100}
//   C[m,k]    = 2|cos(pi u_m/200)| * |sin(pi v/100)/sin(pi v/7600)| * |S|, v=38(k+1)
//   out[b,k,m] = C[m,k]/max(C) * imax*wmax + bias[k,m]   (k<36)
__global__ __launch_bounds__(32) void ocnn_fused_kernel(
    const float* __restrict__ x,       // (16,1,28,28)
    const float* __restrict__ weight,  // (50,36)
    const float* __restrict__ bias,    // (1,36,50)
    float* __restrict__ out,           // (16,36,50)
    float* __restrict__ out_xcnn)      // (16,36,36)
{
  __shared__ float xs[36 * 48];     // x_cnn, zero padded to 36x48
  __shared__ float xh_re[64 * 48];  // Xhat real, later P_re
  __shared__ float xh_im[64 * 48];  // Xhat imag, later P_im
  __shared__ float cmat[64 * 64];   // C magnitudes (valid 50x50)
  __shared__ float tab_c[200];      // cos(pi*n/100)
  __shared__ float tab_s[200];      // sin(pi*n/100)

  const int b = blockIdx.x;
  const int tid = threadIdx.x;
  const int lane16 = tid & 15;
  const int half = tid >> 4;
  const float PI_F = 3.14159265358979323846f;

  // ---- trig LUT (exact integer angle grid) ----
  for (int n = tid; n < 200; n += 32) {
    float s, c;
    sincosf(PI_F * (float)n / 100.0f, &s, &c);
    tab_c[n] = c;
    tab_s[n] = s;
  }
  for (int idx = tid; idx < 36 * 48; idx += 32) xs[idx] = 0.0f;
  __syncthreads();

  // ---- im2col: x_cnn into LDS + second output + imax ----
  float lmax = 0.0f;  // padded region contributes 0
  const float* xb = x + b * 784;
  for (int idx = tid; idx < 36 * 36; idx += 32) {
    int R = idx / 36, Cc = idx % 36;
    int i = R / 6, jj = R % 6, k = Cc / 6, l = Cc % 6;
    int p = 5 * i + k, q = 5 * jj + l;  // index into 31x31 pad(1,2)
    float v = 0.0f;
    if (p >= 1 && p <= 28 && q >= 1 && q <= 28) v = xb[(p - 1) * 28 + (q - 1)];
    xs[R * 48 + Cc] = v;
    out_xcnn[b * 1296 + idx] = v;
    lmax = fmaxf(lmax, v);
  }
  for (int off = 16; off > 0; off >>= 1)
    lmax = fmaxf(lmax, __shfl_xor(lmax, off, 32));
  const float imax = lmax;

  float wl = 0.0f;
  for (int idx = tid; idx < 50 * 36; idx += 32)
    wl = fmaxf(wl, fabsf(weight[idx]));
  for (int off = 16; off > 0; off >>= 1)
    wl = fmaxf(wl, __shfl_xor(wl, off, 32));
  const float wmax = wl;
  __syncthreads();

  // ---- stage 1: Xhat = E1(cos / -sin) x X   (M=64, N=48, K=36) ----
  for (int mt = 0; mt < 4; ++mt) {
    const int mrow = mt * 16 + lane16;  // A-row for this lane
    const int u = 2 * mrow - 50;
    const float mmask = (mrow < 50) ? 1.0f : 0.0f;
    for (int nt = 0; nt < 3; ++nt) {
      v8f acc_re = {0, 0, 0, 0, 0, 0, 0, 0};
      v8f acc_im = {0, 0, 0, 0, 0, 0, 0, 0};
      for (int ks = 0; ks < 9; ++ks) {
        v2f a_c, a_s, bf;
#pragma unroll
        for (int t = 0; t < 2; ++t) {
          const int i = ks * 4 + t + 2 * half;  // K index
          const int n = ((i * u) % 100 + 100) % 100;
          a_c[t] = mmask * tab_c[2 * n];   //  cos(2pi*n/100)
          a_s[t] = -mmask * tab_s[2 * n];  // -sin(2pi*n/100)
          bf[t] = xs[i * 48 + nt * 16 + lane16];
        }
        acc_re = wmma_f32(a_c, bf, acc_re);
        acc_im = wmma_f32(a_s, bf, acc_im);
      }
#pragma unroll
      for (int r = 0; r < 8; ++r) {
        const int m = mt * 16 + r + 8 * half;
        const int j = nt * 16 + lane16;
        xh_re[m * 48 + j] = acc_re[r];
        xh_im[m * 48 + j] = acc_im[r];
      }
    }
  }
  __syncthreads();

  // ---- Hadamard with |W| : P = Xhat .* |W| (in place) ----
  for (int idx = tid; idx < 64 * 48; idx += 32) {
    const int m = idx / 48, j = idx % 48;
    const float w = (m < 50 && j < 36) ? fabsf(weight[m * 36 + j]) : 0.0f;
    xh_re[idx] *= w;
    xh_im[idx] *= w;
  }
  __syncthreads();

  // ---- stage 2: S = P x E2, C = scales * |S|   (M=64, N=64, K=36) ----
  for (int mt = 0; mt < 4; ++mt) {
    for (int nt = 0; nt < 4; ++nt) {
      const int kcol = nt * 16 + lane16;  // output column k (valid < 50)
      v8f s_re = {0, 0, 0, 0, 0, 0, 0, 0};
      v8f s_im = {0, 0, 0, 0, 0, 0, 0, 0};
      for (int ks = 0; ks < 9; ++ks) {
        v2f a_re, a_im, b_c, b_s, b_ns;
#pragma unroll
        for (int t = 0; t < 2; ++t) {
          const int j = ks * 4 + t + 2 * half;  // K index
          a_re[t] = xh_re[(mt * 16 + lane16) * 48 + j];
          a_im[t] = xh_im[(mt * 16 + lane16) * 48 + j];
          const int n = (j * 38 * (kcol + 1)) % 100;
          b_c[t] = tab_c[2 * n];
          b_s[t] = tab_s[2 * n];
          b_ns[t] = -b_s[t];
        }
        // (Pre + i Pim)(c - i s) -> re: Pre*c + Pim*s ; im: Pim*c - Pre*s
        s_re = wmma_f32(a_re, b_c, s_re);
        s_re = wmma_f32(a_im, b_s, s_re);
        s_im = wmma_f32(a_im, b_c, s_im);
        s_im = wmma_f32(a_re, b_ns, s_im);
      }
      // column scale |G(v)| = |sin(pi*v/100)/sin(pi*v/7600)|, v = 38(k+1)
      const int v38 = 38 * (kcol + 1);
      const float num = tab_s[v38 % 200];
      const float den = sinf(PI_F * (float)v38 / 7600.0f);
      const float Bk = fabsf(num / den);
#pragma unroll
      for (int r = 0; r < 8; ++r) {
        const int m = mt * 16 + r + 8 * half;
        const int nAm = ((m - 25) % 200 + 200) % 200;  // cos(pi*(m-25)/100)
        const float Am = 2.0f * fabsf(tab_c[nAm]);
        const float mag = sqrtf(s_re[r] * s_re[r] + s_im[r] * s_im[r]);
        cmat[m * 64 + kcol] = Am * Bk * mag;
      }
    }
  }
  __syncthreads();

  // ---- per-batch max over valid 50x50, then emit out[b,k,m] ----
  float cl = 0.0f;
  for (int idx = tid; idx < 2500; idx += 32) {
    const int m = idx / 50, k = idx % 50;
    cl = fmaxf(cl, cmat[m * 64 + k]);
  }
  for (int off = 16; off > 0; off >>= 1)
    cl = fmaxf(cl, __shfl_xor(cl, off, 32));
  const float scale = imax * wmax / cl;

  for (int idx = tid; idx < 36 * 50; idx += 32) {
    const int k = idx / 50, m = idx % 50;
    out[b * 1800 + idx] = cmat[m * 64 + k] * scale + bias[idx];
  }
}

extern "C" void kernel_launch(void* const* d_in, const int* in_sizes, int n_in,
                              void* d_out, int out_size, void* d_ws,
                              size_t ws_size, hipStream_t stream) {
  const float* x = (const float*)d_in[0];       // (16,1,28,28)
  const float* weight = (const float*)d_in[1];  // (50,36)
  const float* bias = (const float*)d_in[2];    // (1,36,50)
  float* out = (float*)d_out;                   // 16*36*50 floats
  float* out_xcnn = out + 16 * 36 * 50;         // 16*36*36 floats
  (void)in_sizes; (void)n_in; (void)out_size; (void)d_ws; (void)ws_size;
  ocnn_fused_kernel<<<16, 32, 0, stream>>>(x, weight, bias, out, out_xcnn);
}